// GCN_B_50448685859072
// MI455X (gfx1250) — compile-verified
//
#include <hip/hip_runtime.h>
#include <math.h>

typedef float v2f __attribute__((ext_vector_type(2)));
typedef float v8f __attribute__((ext_vector_type(8)));

// ---------------------------------------------------------------------------
// Utility kernels
// ---------------------------------------------------------------------------
__global__ void zerof_kernel(float* __restrict__ p, long n) {
  long i = (long)blockIdx.x * blockDim.x + threadIdx.x;
  long stride = (long)gridDim.x * blockDim.x;
  for (; i < n; i += stride) p[i] = 0.0f;
}

__global__ void degree_kernel(const int* __restrict__ src, const int* __restrict__ dst,
                              float* __restrict__ deg_out, float* __restrict__ deg_in,
                              int E) {
  int i = blockIdx.x * blockDim.x + threadIdx.x;
  if (i < E) {
    atomicAdd(&deg_out[src[i]], 1.0f);
    atomicAdd(&deg_in[dst[i]], 1.0f);
  }
}

__global__ void norm_kernel(const float* __restrict__ deg_in, const float* __restrict__ deg_out,
                            float* __restrict__ isi, float* __restrict__ iso,
                            float* __restrict__ nbuf, int N) {
  int i = blockIdx.x * blockDim.x + threadIdx.x;
  if (i < N) {
    float di = fmaxf(deg_in[i], 1.0f);
    float dq = fmaxf(deg_out[i], 1.0f);
    isi[i]  = 1.0f / sqrtf(di);
    iso[i]  = 1.0f / sqrtf(dq);
    nbuf[i] = 1.0f / di;
  }
}

// Edge scatter, one wave32 per edge per iteration: gather the 128-float row of
// h[src] (float4/lane), scale by iso[src], atomically accumulate into agg[dst]
// (segment_sum semantics).  Grid-stride over edges with software prefetch of
// the NEXT edge's source row (global_prefetch_b8) so the random L2 gather
// latency overlaps with the current edge's atomic issue.
__global__ void scatter_kernel(const float* __restrict__ h, int ldh,
                               const int* __restrict__ src, const int* __restrict__ dst,
                               const float* __restrict__ iso, float* __restrict__ agg,
                               int E) {
  const int lane = threadIdx.x & 31;
  const int nwaves = (gridDim.x * blockDim.x) >> 5;
  int e = blockIdx.x * (blockDim.x >> 5) + (threadIdx.x >> 5);
  for (; e < E; e += nwaves) {
    const int en = e + nwaves;
    if (en < E) {
      const int sn = src[en];
      __builtin_prefetch(h + (size_t)sn * ldh + lane * 4, 0, 3);
    }
    const int s = src[e];
    const int d = dst[e];
    const float sc = iso[s];
    float4 v = *(const float4*)(h + (size_t)s * ldh + lane * 4);
    float* ap = agg + (size_t)d * 128 + lane * 4;
    atomicAdd(ap + 0, v.x * sc);
    atomicAdd(ap + 1, v.y * sc);
    atomicAdd(ap + 2, v.z * sc);
    atomicAdd(ap + 3, v.w * sc);
  }
}

// ---------------------------------------------------------------------------
// Fused per-layer GEMM:
//   out = [agg*isi | nbuf*feat | nbuf*Hcat] @ [W ; Wb] + b + nbuf*bb  (ReLU opt)
// WMMA fp32 16x16x4. 256 threads = 8 waves; block tile 128(M) x 128(N),
// K staged through LDS in chunks of 16.
//
// B is stored in LDS in WMMA-fragment-native pair layout:
//   Bl[kpair * LDBP + col*2 + {0,1}] = B[2*kpair + {0,1}][col]
// so each lane's B fragment is one contiguous, 8B-aligned ds_load_b64.
// LDBP = 288 dwords (== 32 mod 64) => lanes 0-15 (kpair even) and lanes
// 16-31 (kpair odd) hit disjoint 32-bank halves: conflict-free b64 reads.
// ---------------------------------------------------------------------------
#define MB    128
#define KC    16
#define LDA   20     // padded LDS row stride for A (floats)
#define LDBP  288    // LDS stride per k-pair row of B (floats)

__global__ __launch_bounds__(256) void gemm_fused_kernel(
    const float* __restrict__ agg,   // [N,128]
    const float* __restrict__ feat,  // [N,128]
    const float* __restrict__ Hcat,  // [N,256] = h1|h2
    const float* __restrict__ isi,   // [N]
    const float* __restrict__ nbuf,  // [N]
    const float* __restrict__ W,     // [128,128] row-major
    const float* __restrict__ Wb,    // [Ktot-128,128] row-major
    const float* __restrict__ bv_,   // [128]
    const float* __restrict__ bbv_,  // [128]
    float* __restrict__ out, int ldo,
    int N, int Ktot, int relu)
{
  __shared__ float Al[MB * LDA];          // 10240 B
  __shared__ float Bl[(KC / 2) * LDBP];   //  9216 B

  const int tid   = threadIdx.x;
  const int r0    = blockIdx.x * MB;
  const int lane  = tid & 31;
  const int wave  = tid >> 5;
  const int m     = lane & 15;   // row (A frag) / col (B frag) within 16
  const int khalf = lane >> 4;   // lanes 0-15: K{0,1}; lanes 16-31: K{2,3}

  v8f acc[8];
#pragma unroll
  for (int t = 0; t < 8; ++t)
#pragma unroll
    for (int j = 0; j < 8; ++j) acc[t][j] = 0.0f;

  for (int k0 = 0; k0 < Ktot; k0 += KC) {
    // ---- region select (chunk lies fully inside one region: boundaries at 128/256)
    const float* aptr; int ald; const float* sc;
    if (k0 < 128)      { aptr = agg  + k0;         ald = 128; sc = isi;  }
    else if (k0 < 256) { aptr = feat + (k0 - 128); ald = 128; sc = nbuf; }
    else               { aptr = Hcat + (k0 - 256); ald = 256; sc = nbuf; }

    // ---- stage A tile: 128 rows x 16 k, scaled per row
#pragma unroll
    for (int i = 0; i < 2; ++i) {
      int f = tid + i * 256;            // 0..511 -> 512 float4
      int row = f >> 2, q = f & 3;
      int r = r0 + row;
      float4 v; v.x = v.y = v.z = v.w = 0.0f;
      if (r < N) {
        v = *(const float4*)(aptr + (size_t)r * ald + q * 4);
        float s = sc[r];
        v.x *= s; v.y *= s; v.z *= s; v.w *= s;
      }
      *(float4*)&Al[row * LDA + q * 4] = v;
    }

    // ---- stage B tile (16 k x 128 n) into pair-interleaved layout
    const float* bptr = (k0 < 128) ? (W + (size_t)k0 * 128)
                                   : (Wb + (size_t)(k0 - 128) * 128);
#pragma unroll
    for (int i = 0; i < 4; ++i) {
      int f = tid + i * 256;            // 0..1023 -> 8 kpairs x 128 cols
      int kpair = f >> 7;
      int col = f & 127;
      v2f w;
      w.x = bptr[(size_t)(2 * kpair)     * 128 + col];
      w.y = bptr[(size_t)(2 * kpair + 1) * 128 + col];
      *(v2f*)&Bl[kpair * LDBP + col * 2] = w;
    }
    __syncthreads();

    // ---- 4 WMMA k-steps x 8 N-tiles
    const float* Arow = &Al[(wave * 16 + m) * LDA];
#pragma unroll
    for (int ks = 0; ks < 4; ++ks) {
      const int kb = ks * 4 + khalf * 2;       // even
      v2f a = *(const v2f*)(Arow + kb);        // contiguous b64
      const float* bp = &Bl[(ks * 2 + khalf) * LDBP + m * 2];
#pragma unroll
      for (int t = 0; t < 8; ++t) {
        v2f bf = *(const v2f*)(bp + t * 32);   // contiguous b64
        acc[t] = __builtin_amdgcn_wmma_f32_16x16x4_f32(
            false, a, false, bf, (short)0, acc[t], false, false);
      }
    }
    __syncthreads();
  }

  // ---- epilogue: + b + nbuf*bb, optional ReLU, store
  // D layout: element (vgpr j, lane) -> row = j + 8*khalf, col = lane%16 + 16*t
  const int rbase = r0 + wave * 16 + khalf * 8;
  float nb[8];
#pragma unroll
  for (int j = 0; j < 8; ++j) nb[j] = (rbase + j < N) ? nbuf[rbase + j] : 0.0f;

#pragma unroll
  for (int t = 0; t < 8; ++t) {
    const int col = t * 16 + m;
    const float bv  = bv_[col];
    const float bbv = bbv_[col];
#pragma unroll
    for (int j = 0; j < 8; ++j) {
      const int r = rbase + j;
      if (r < N) {
        float v = acc[t][j] + bv + nb[j] * bbv;
        if (relu) v = fmaxf(v, 0.0f);
        out[(size_t)r * ldo + col] = v;
      }
    }
  }
}

// ---------------------------------------------------------------------------
// Launch
// ---------------------------------------------------------------------------
extern "C" void kernel_launch(void* const* d_in, const int* in_sizes, int n_in,
                              void* d_out, int out_size, void* d_ws, size_t ws_size,
                              hipStream_t stream) {
  const float* feat = (const float*)d_in[0];
  const int*   src  = (const int*)d_in[1];
  const int*   dst  = (const int*)d_in[2];
  const float* W[3]   = {(const float*)d_in[3],  (const float*)d_in[7],  (const float*)d_in[11]};
  const float* bW[3]  = {(const float*)d_in[4],  (const float*)d_in[8],  (const float*)d_in[12]};
  const float* Wbf[3] = {(const float*)d_in[5],  (const float*)d_in[9],  (const float*)d_in[13]};
  const float* bbf[3] = {(const float*)d_in[6],  (const float*)d_in[10], (const float*)d_in[14]};

  const int N = in_sizes[0] / 128;
  const int E = in_sizes[1];

  char* ws = (char*)d_ws;
  size_t off = 0;
#define CARVE(T, name, count)                                           \
  T* name = (T*)(ws + off);                                             \
  off = (off + (size_t)(count) * sizeof(T) + 255) & ~(size_t)255;

  CARVE(float, deg_in,  N);
  CARVE(float, deg_out, N);
  CARVE(float, isi,     N);
  CARVE(float, iso,     N);
  CARVE(float, nbuf,    N);
  CARVE(float, agg,     (size_t)N * 128);
  CARVE(float, Hbuf,    (size_t)N * 256);
#undef CARVE
  (void)ws_size; (void)n_in;

  // degrees + norms
  zerof_kernel<<<512, 256, 0, stream>>>(deg_in,  (long)N);
  zerof_kernel<<<512, 256, 0, stream>>>(deg_out, (long)N);
  degree_kernel<<<(E + 255) / 256, 256, 0, stream>>>(src, dst, deg_out, deg_in, E);
  norm_kernel<<<(N + 255) / 256, 256, 0, stream>>>(deg_in, deg_out, isi, iso, nbuf, N);

  const int   Ktot[3] = {256, 384, 512};
  const float* hsrc[3] = {feat, Hbuf, Hbuf + 128};
  const int   ldh[3]  = {128, 256, 256};
  float*      outp[3] = {Hbuf, Hbuf + 128, (float*)d_out};
  const int   ldo[3]  = {256, 256, 128};

  // ~2 edges per wave -> the prefetch of iteration e+nwaves has work to hide
  const int scatter_blocks = (E / 2 + 7) / 8;

  for (int l = 0; l < 3; ++l) {
    zerof_kernel<<<2048, 256, 0, stream>>>(agg, (long)N * 128);
    scatter_kernel<<<scatter_blocks, 256, 0, stream>>>(hsrc[l], ldh[l], src, dst, iso, agg, E);
    gemm_fused_kernel<<<(N + MB - 1) / MB, 256, 0, stream>>>(
        agg, feat, Hbuf, isi, nbuf,
        W[l], Wbf[l], bW[l], bbf[l],
        outp[l], ldo[l], N, Ktot[l], (l != 2) ? 1 : 0);
  }
}